// L1Loss_weight_jyoukenn_89910845375274
// MI455X (gfx1250) — compile-verified
//
#include <hip/hip_runtime.h>
#include <hip/hip_bf16.h>

// MI455X (gfx1250): HBM-bound streaming reduction (floor ~14.4us @ 23.3TB/s).
//   Pass 1: 1024 blocks x 256 thr, grid-stride float4 loads (5 streams,
//           20B/elem, 335MB), 2x unrolled => 10 b128 loads in flight/wave,
//           per-wave LDS 64-bin ds_add_f32, wave32 shfl total,
//           per-block 65-float partials -> d_ws.
//   Pass 2: column-sum of the (NB x 65) partial matrix via
//           V_WMMA_F32_16X16X4_F32 (ones-vector matmul, full f32).
//           Loads are unconditional with a clamped base pointer: invalid
//           lanes only pollute their own (discarded) columns, so no
//           predication is needed -> branch-free inner loop.
// Deterministic: no global atomics; d_out fully rewritten each call.

typedef float f32x4 __attribute__((ext_vector_type(4)));
typedef int   i32x4 __attribute__((ext_vector_type(4)));
typedef float v2f   __attribute__((ext_vector_type(2)));
typedef float v8f   __attribute__((ext_vector_type(8)));

#define C_MULT   2.0f
#define OVER_LIM 0.5f

__device__ __forceinline__ float elem_x(float o, float y, float g, float w) {
    float x  = fabsf(y - o);
    float xo = (fabsf(y) - fabsf(o) > 0.0f) ? x : x * C_MULT;
    x = (fabsf(g) > OVER_LIM) ? xo : x;
    return x * w;
}

__device__ __forceinline__ float process4(
    const f32x4* __restrict__ o4, const f32x4* __restrict__ y4,
    const f32x4* __restrict__ g4, const f32x4* __restrict__ w4,
    const i32x4* __restrict__ l4, int i, float* __restrict__ mybins)
{
    f32x4 o = __builtin_nontemporal_load(o4 + i);
    f32x4 y = __builtin_nontemporal_load(y4 + i);
    f32x4 g = __builtin_nontemporal_load(g4 + i);
    f32x4 w = __builtin_nontemporal_load(w4 + i);
    i32x4 lb = __builtin_nontemporal_load(l4 + i);
    float t = 0.0f;
#pragma unroll
    for (int j = 0; j < 4; ++j) {
        float x = elem_x(o[j], y[j], g[j], w[j]);
        t += x;
        atomicAdd(&mybins[lb[j] & 63], x);   // ds_add_f32
    }
    return t;
}

__global__ __launch_bounds__(256) void l1seg_pass1(
    const float* __restrict__ outs, const float* __restrict__ y1,
    const float* __restrict__ wlab, const float* __restrict__ wt,
    const int*   __restrict__ lab,  float* __restrict__ ws, int n)
{
    __shared__ float bins[8][64];   // per-wave bin replication (8 waves)
    __shared__ float wtot[8];

    const int tid  = threadIdx.x;
    const int wave = tid >> 5;
    const int lane = tid & 31;

    for (int i = tid; i < 8 * 64; i += 256) ((float*)bins)[i] = 0.0f;
    __syncthreads();

    const f32x4* o4 = (const f32x4*)outs;
    const f32x4* y4 = (const f32x4*)y1;
    const f32x4* g4 = (const f32x4*)wlab;
    const f32x4* w4 = (const f32x4*)wt;
    const i32x4* l4 = (const i32x4*)lab;
    float* mybins = bins[wave];

    const int nvec   = n >> 2;
    const int stride = (int)gridDim.x * 256;
    float tot = 0.0f;

    // 2x-unrolled grid-stride: 10 b128 loads in flight per wave
    int i = (int)blockIdx.x * 256 + tid;
    for (; i + stride < nvec; i += 2 * stride) {
        tot += process4(o4, y4, g4, w4, l4, i,          mybins);
        tot += process4(o4, y4, g4, w4, l4, i + stride, mybins);
    }
    if (i < nvec)
        tot += process4(o4, y4, g4, w4, l4, i, mybins);

    // tail (n % 4), handled by block 0 only
    if (blockIdx.x == 0) {
        int idx = (nvec << 2) + tid;
        if (idx < n) {
            float x = elem_x(outs[idx], y1[idx], wlab[idx], wt[idx]);
            tot += x;
            atomicAdd(&mybins[lab[idx] & 63], x);
        }
    }

    // wave32 reduction of the running total
#pragma unroll
    for (int off = 16; off > 0; off >>= 1)
        tot += __shfl_down(tot, off, 32);
    if (lane == 0) wtot[wave] = tot;
    __syncthreads();

    float* dst = ws + (size_t)blockIdx.x * 65;
    if (tid < 64) {
        float s = 0.0f;
#pragma unroll
        for (int w2 = 0; w2 < 8; ++w2) s += bins[w2][tid];
        dst[tid] = s;                 // cols 0..63 = label bins
    } else if (tid == 64) {
        float s = 0.0f;
#pragma unroll
        for (int w2 = 0; w2 < 8; ++w2) s += wtot[w2];
        dst[64] = s;                  // col 64 = total
    }
}

// Column-sum of ws (rows x 65) via WMMA: D = ones(16x4) * B(4x16) + C.
// A == all-ones makes the result independent of the k<->(vgpr,half)
// operand mapping; every row of D equals the column sum, so any lane's
// c[0] is the sum for column (tile*16 + lane%16). Lanes whose column
// is out of range (tile 4, ncol>0) load from a clamped pointer (always
// in-bounds) and only corrupt columns that are never stored.
__global__ __launch_bounds__(160) void l1seg_pass2(
    const float* __restrict__ ws, float* __restrict__ out,
    int rows, float invN)
{
    const int tid  = threadIdx.x;
    const int tile = tid >> 5;        // 5 waves -> 5 column tiles of 16
    const int lane = tid & 31;
    const int ncol = lane & 15;
    const int half = lane >> 4;
    const int col  = tile * 16 + ncol;
    const float* p = ws + ((col < 65) ? col : 64);   // clamped, in-bounds

    v2f a; a[0] = 1.0f; a[1] = 1.0f;
    v8f c0 = {}, c1 = {}, c2 = {}, c3 = {};

    for (int kb = 0; kb < rows; kb += 16) {
        const size_t r = (size_t)(kb + half) * 65;
        v2f b0, b1, b2, b3;
        b0[0] = p[r      ];  b0[1] = p[r + 130];   // rows kb+half, kb+2+half
        b1[0] = p[r + 260];  b1[1] = p[r + 390];   // +4, +6
        b2[0] = p[r + 520];  b2[1] = p[r + 650];   // +8, +10
        b3[0] = p[r + 780];  b3[1] = p[r + 910];   // +12, +14
        c0 = __builtin_amdgcn_wmma_f32_16x16x4_f32(false, a, false, b0, (short)0, c0, false, false);
        c1 = __builtin_amdgcn_wmma_f32_16x16x4_f32(false, a, false, b1, (short)0, c1, false, false);
        c2 = __builtin_amdgcn_wmma_f32_16x16x4_f32(false, a, false, b2, (short)0, c2, false, false);
        c3 = __builtin_amdgcn_wmma_f32_16x16x4_f32(false, a, false, b3, (short)0, c3, false, false);
    }

    v8f c = (c0 + c1) + (c2 + c3);
    float colsum = c[0];
    if (lane < 16 && col < 65) {
        // out[0] = loss (ws col 64); out[1+u] = loss_array[u] (ws col u)
        out[(col == 64) ? 0 : (col + 1)] = colsum * invN;
    }
}

extern "C" void kernel_launch(void* const* d_in, const int* in_sizes, int n_in,
                              void* d_out, int out_size, void* d_ws, size_t ws_size,
                              hipStream_t stream) {
    const float* outs = (const float*)d_in[0];
    const float* y1   = (const float*)d_in[1];
    const float* wlab = (const float*)d_in[2];
    const float* wt   = (const float*)d_in[3];
    const int*   lab  = (const int*)d_in[4];
    float* out = (float*)d_out;
    float* ws  = (float*)d_ws;
    const int n = in_sizes[0];

    int NB = 1024;                                  // pass-1 blocks (rows in ws)
    size_t need = (size_t)NB * 65 * sizeof(float);
    if (ws_size < need) {                           // defensive clamp
        NB = (int)(ws_size / (65 * sizeof(float)));
        NB &= ~15;                                  // pass-2 needs rows % 16 == 0
        if (NB < 16) NB = 16;
    }

    hipLaunchKernelGGL(l1seg_pass1, dim3(NB), dim3(256), 0, stream,
                       outs, y1, wlab, wt, lab, ws, n);
    hipLaunchKernelGGL(l1seg_pass2, dim3(1), dim3(160), 0, stream,
                       ws, out, NB, 1.0f / (float)n);
}